// GraphSumEmbedding_62921270886789
// MI455X (gfx1250) — compile-verified
//
#include <hip/hip_runtime.h>

// GraphSumEmbedding for MI455X (gfx1250, wave32).
// Pipeline: zero ws -> scatter1 (atomics) -> layer1 (WMMA f32 + BN + ReLU)
//           -> scatter2 -> layer2 (WMMA f32).

typedef __attribute__((ext_vector_type(2))) float v2f;
typedef __attribute__((ext_vector_type(8))) float v8f;

#define DFEAT 64
#define BN_EPS 1e-5f

// ---------------------------------------------------------------- zero ws
__global__ __launch_bounds__(256) void zero_kernel(float4* __restrict__ p, int n4) {
    int i = blockIdx.x * blockDim.x + threadIdx.x;
    if (i < n4) p[i] = make_float4(0.f, 0.f, 0.f, 0.f);
}

// ------------------------------------------------- edge scatter-add (sum agg)
// 16 threads per edge, each handling 4 consecutive features (float4 gather,
// 4 f32 atomics). x/h and agg are L2-resident (25.6MB each vs 192MB L2).
__global__ __launch_bounds__(256) void scatter_kernel(
    const float* __restrict__ feat, const int* __restrict__ src,
    const int* __restrict__ dst, float* __restrict__ agg, int nE) {
    unsigned tid = blockIdx.x * blockDim.x + threadIdx.x;
    int e = tid >> 4;
    int q = (tid & 15) << 2;
    if (e < nE) {
        int s = src[e];
        int d = dst[e];
        float4 v = *(const float4*)(feat + (size_t)s * DFEAT + q);
        float* p = agg + (size_t)d * DFEAT + q;
        atomicAdd(p + 0, v.x);
        atomicAdd(p + 1, v.y);
        atomicAdd(p + 2, v.z);
        atomicAdd(p + 3, v.w);
    }
}

// ------------------------------------- fused: agg@W_rel + b + feat@W_root,
//                                       optional BN(eval) + ReLU epilogue.
// Block = 256 threads = 8 waves; each wave computes a 16-node x 64-col tile
// with four 16x16 f32 accumulators, K stepped by 4 via V_WMMA_F32_16X16X4_F32.
__global__ __launch_bounds__(256) void layer_kernel(
    const float* __restrict__ agg, const float* __restrict__ feat,
    const float* __restrict__ W_rel, const float* __restrict__ b_rel,
    const float* __restrict__ W_root,
    const float* __restrict__ gamma, const float* __restrict__ beta,
    const float* __restrict__ mean, const float* __restrict__ var,
    float* __restrict__ out, int n_nodes, int bn_relu) {
    __shared__ float sWrel[DFEAT * DFEAT];
    __shared__ float sWroot[DFEAT * DFEAT];

    int tid = threadIdx.x;
    // Stage both 64x64 f32 weight matrices into LDS (16KB each).
    {
        const float4* s0 = (const float4*)W_rel;
        const float4* s1 = (const float4*)W_root;
        float4* d0 = (float4*)sWrel;
        float4* d1 = (float4*)sWroot;
#pragma unroll
        for (int i = 0; i < 4; ++i) {
            d0[tid + 256 * i] = s0[tid + 256 * i];
            d1[tid + 256 * i] = s1[tid + 256 * i];
        }
    }
    __syncthreads();

    int wave = tid >> 5;
    int lane = tid & 31;
    int col = lane & 15;            // B/C/D column within 16-wide tile
    int koff = (lane >> 4) << 1;    // f32 A/B: upper half-wave holds K+2,K+3
    int rhalf = (lane >> 4) << 3;   // C/D: upper half-wave holds rows M+8..M+15

    int node0 = blockIdx.x * 128 + wave * 16;
    int arow = node0 + col;                       // A-matrix row (M = lane&15)
    if (arow >= n_nodes) arow = n_nodes - 1;      // clamp: keep EXEC all-1s

    // Accumulators seeded with the bias (broadcast down each column).
    v8f acc[4];
#pragma unroll
    for (int t = 0; t < 4; ++t) {
        float bv = b_rel[t * 16 + col];
#pragma unroll
        for (int g = 0; g < 8; ++g) acc[t][g] = bv;
    }

    const float* arow_agg = agg + (size_t)arow * DFEAT;
    const float* arow_feat = feat + (size_t)arow * DFEAT;

#pragma unroll
    for (int pass = 0; pass < 2; ++pass) {
        const float* A = (pass == 0) ? arow_agg : arow_feat;
        const float* W = (pass == 0) ? sWrel : sWroot;
#pragma unroll
        for (int k = 0; k < DFEAT; k += 4) {
            v2f a;
            a.x = A[k + koff];
            a.y = A[k + koff + 1];
#pragma unroll
            for (int t = 0; t < 4; ++t) {
                v2f b;
                b.x = W[(k + koff) * DFEAT + t * 16 + col];
                b.y = W[(k + koff + 1) * DFEAT + t * 16 + col];
                acc[t] = __builtin_amdgcn_wmma_f32_16x16x4_f32(
                    false, a, false, b, (short)0, acc[t], false, false);
            }
        }
    }

    // Epilogue: optional BatchNorm(eval) + ReLU, then scatter rows to memory.
#pragma unroll
    for (int t = 0; t < 4; ++t) {
        int n = t * 16 + col;
        float scale = 1.f, shift = 0.f;
        if (bn_relu) {
            scale = gamma[n] * rsqrtf(var[n] + BN_EPS);
            shift = beta[n] - mean[n] * scale;
        }
#pragma unroll
        for (int g = 0; g < 8; ++g) {
            int row = node0 + rhalf + g;
            if (row < n_nodes) {
                float v = acc[t][g] * scale + shift;
                if (bn_relu) v = fmaxf(v, 0.f);
                out[(size_t)row * DFEAT + n] = v;
            }
        }
    }
}

extern "C" void kernel_launch(void* const* d_in, const int* in_sizes, int n_in,
                              void* d_out, int out_size, void* d_ws, size_t ws_size,
                              hipStream_t stream) {
    const float* x = (const float*)d_in[0];
    const int* edge_index = (const int*)d_in[2];
    const float* W1_rel = (const float*)d_in[5];
    const float* b1_rel = (const float*)d_in[6];
    const float* W1_root = (const float*)d_in[7];
    const float* gamma = (const float*)d_in[8];
    const float* beta = (const float*)d_in[9];
    const float* mean = (const float*)d_in[10];
    const float* var = (const float*)d_in[11];
    const float* W2_rel = (const float*)d_in[12];
    const float* b2_rel = (const float*)d_in[13];
    const float* W2_root = (const float*)d_in[14];

    int nN = in_sizes[0] / DFEAT;   // 100000
    int nE = in_sizes[2] / 2;       // 1600000
    const int* src = edge_index;
    const int* dst = edge_index + nE;

    float* agg1 = (float*)d_ws;
    float* h = agg1 + (size_t)nN * DFEAT;
    float* agg2 = h + (size_t)nN * DFEAT;
    float* out = (float*)d_out;

    // 1) zero agg1 / h / agg2
    int n4 = (3 * nN * DFEAT) / 4;
    zero_kernel<<<(n4 + 255) / 256, 256, 0, stream>>>((float4*)d_ws, n4);

    // 2) agg1[dst] += x[src]
    unsigned sthreads = (unsigned)nE * 16u;
    scatter_kernel<<<(sthreads + 255) / 256, 256, 0, stream>>>(x, src, dst, agg1, nE);

    // 3) h = relu(bn(agg1@W1_rel + b1 + x@W1_root))
    int lblocks = (nN + 127) / 128;
    layer_kernel<<<lblocks, 256, 0, stream>>>(agg1, x, W1_rel, b1_rel, W1_root,
                                              gamma, beta, mean, var, h, nN, 1);

    // 4) agg2[dst] += h[src]
    scatter_kernel<<<(sthreads + 255) / 256, 256, 0, stream>>>(h, src, dst, agg2, nE);

    // 5) out = agg2@W2_rel + b2 + h@W2_root
    layer_kernel<<<lblocks, 256, 0, stream>>>(agg2, h, W2_rel, b2_rel, W2_root,
                                              gamma, beta, mean, var, out, nN, 0);
}